// SSIM_L1Loss_55293408969262
// MI455X (gfx1250) — compile-verified
//
#include <hip/hip_runtime.h>

// SSIM3D(window=11) + SmoothL1 fused loss for [2,1,160,160,160] fp32 on gfx1250.
// Box filters = banded-matrix matmuls on V_WMMA_F32_16X16X4_F32.
// Tile movement via Tensor Data Mover (tensor_load_to_lds / tensor_store_from_lds).

typedef __attribute__((ext_vector_type(2))) float v2f;
typedef __attribute__((ext_vector_type(8))) float v8f;
typedef unsigned int v4u __attribute__((ext_vector_type(4)));
typedef int v4i __attribute__((ext_vector_type(4)));
typedef int v8i __attribute__((ext_vector_type(8)));

#define WSZ   160
#define HW    25600      // 160*160
#define BVOL  4096000    // 160^3
#define NT    8192000    // 2*160^3 elements per channel
#define NPART 3200       // l1 partials [0,3200), ssim partials [3200,6400)

__device__ __forceinline__ float wave_sum(float v) {
#pragma unroll
  for (int off = 16; off > 0; off >>= 1) v += __shfl_xor(v, off, 32);
  return v;
}

// ---- Tensor Data Mover 2D tile descriptors ------------------------------
// D# group 0 (ISA 08 §8.3): count=1, lds_addr[63:32], global_addr[120:64], type=2.
__device__ __forceinline__ v4u tdm_g0(void* lds, const void* gmem) {
  const unsigned lds_off = (unsigned)(unsigned long long)lds; // low 32b = LDS byte offset
  const unsigned long long ga = (unsigned long long)gmem;
  v4u g0;
  g0.x = 1u;                                                  // count=1, user mode
  g0.y = lds_off;
  g0.z = (unsigned)ga;                                        // global_addr[31:0]
  g0.w = (unsigned)((ga >> 32) & 0x01FFFFFFu) | (2u << 30);   // [56:32] | type=2
  return g0;
}

// D# group 1 (§8.4): data_size=4B; tensor_dims huge (no OOB clip);
// tile_dim0 = contiguous elems per line, tile_dim1 = #lines, dim0 stride in elems.
__device__ __forceinline__ v8i tdm_g1(int tile0, int tile1, int stride_elems) {
  const unsigned td0 = 1u << 30, td1 = 1u << 30;
  v8i g1;
  g1[0] = (int)(2u << 16);                                           // data_size=4B
  g1[1] = (int)((td0 & 0xFFFFu) << 16);                              // tensor_dim0 lo
  g1[2] = (int)(((td0 >> 16) & 0xFFFFu) | ((td1 & 0xFFFFu) << 16));  // td0 hi | td1 lo
  g1[3] = (int)(((td1 >> 16) & 0xFFFFu) |
                (((unsigned)tile0 & 0xFFFFu) << 16));                // td1 hi | tile_dim0
  g1[4] = (int)((unsigned)tile1 & 0xFFFFu);                          // tile_dim1
  g1[5] = stride_elems;                                              // tensor_dim0_stride
  g1[6] = 0;
  g1[7] = 0;
  return g1;
}

__device__ __forceinline__ void tdm_load_2d(void* lds, const void* gmem,
                                            int tile0, int tile1, int stride) {
  v4u g0 = tdm_g0(lds, gmem);
  v8i g1 = tdm_g1(tile0, tile1, stride);
  v4i z4 = (v4i)0;
#if defined(__clang_major__) && (__clang_major__ >= 23)
  v8i z8 = (v8i)0;
  __builtin_amdgcn_tensor_load_to_lds(g0, g1, z4, z4, z8, 0);
#else
  __builtin_amdgcn_tensor_load_to_lds(g0, g1, z4, z4, 0);
#endif
}

__device__ __forceinline__ void tdm_store_2d(void* lds, void* gmem,
                                             int tile0, int tile1, int stride) {
  v4u g0 = tdm_g0(lds, gmem);
  v8i g1 = tdm_g1(tile0, tile1, stride);
  v4i z4 = (v4i)0;
#if defined(__clang_major__) && (__clang_major__ >= 23)
  v8i z8 = (v8i)0;
  __builtin_amdgcn_tensor_store_from_lds(g0, g1, z4, z4, z8, 0);
#else
  __builtin_amdgcn_tensor_store_from_lds(g0, g1, z4, z4, 0);
#endif
}

// ---- WMMA fp32 box-sum of one 16-row output tile ------------------------
// t: 160x16 tile in LDS, row-major t[r*16 + p].
// D[m][n] = sum_{|m-k|<=5} t[k][n], m in [m0, m0+16).
// A-frag (32-bit 16x4): lanes 0-15 -> M, VGPR/half -> K pairs; B mirrored; C/D per ISA.
__device__ __forceinline__ v8f box_tile(const float* __restrict__ t, int m0) {
  const int lane = threadIdx.x & 31;
  const int half = lane >> 4;
  const int n    = lane & 15;
  const int m    = m0 + (lane & 15);
  v8f acc = {};
  const int kt0 = (m0 >= 5) ? ((m0 - 5) >> 2) : 0;
  const int kt1 = ((m0 + 20) <= 159) ? ((m0 + 20) >> 2) : 39;
  for (int kt = kt0; kt <= kt1; ++kt) {
    const int ka = (kt << 2) + half * 2;     // this lane's two K indices
    v2f a, b;
    const int d0 = m - ka, d1 = m - ka - 1;
    a.x = (d0 >= -5 && d0 <= 5) ? 1.0f : 0.0f;
    a.y = (d1 >= -5 && d1 <= 5) ? 1.0f : 0.0f;
    b.x = t[ka * 16 + n];
    b.y = t[ka * 16 + 16 + n];
    acc = __builtin_amdgcn_wmma_f32_16x16x4_f32(false, a, false, b,
                                                (short)0, acc, false, false);
  }
  return acc;
}

// ---- Pass 1: build 5 channels, W-axis box sum, smoothL1 partials --------
// group = 16 H-pencils at fixed (b,d) slab; 3200 single-wave blocks.
__global__ void __launch_bounds__(32) k_wpass(const float* __restrict__ x,
                                              const float* __restrict__ y,
                                              float* __restrict__ ws) {
  __shared__ float ch[5][2560];       // channel tiles, ch[c][w*16 + p]
  __shared__ float outb[2560];        // staging: outb[n*160 + m] (lines of 160)
  const int lane = threadIdx.x & 31;
  const int g    = blockIdx.x;                    // 0..3199
  const int slab = g / 10;                        // b*160 + d
  const int h0   = (g % 10) * 16;
  const long base = (long)slab * HW + (long)h0 * WSZ;

  float l1 = 0.f;
  for (int p = 0; p < 16; ++p) {                  // coalesced row reads
    const long rb = base + (long)p * WSZ;
    for (int w = lane; w < WSZ; w += 32) {
      const float xv = x[rb + w], yv = y[rb + w];
      const int li = w * 16 + p;
      ch[0][li] = xv;
      ch[1][li] = yv;
      ch[2][li] = xv * xv;
      ch[3][li] = yv * yv;
      ch[4][li] = xv * yv;
      const float d = fabsf(xv - yv);
      l1 += (d < 1.f) ? 0.5f * d * d : (d - 0.5f);   // beta = 1
    }
  }
  const float wsum = wave_sum(l1);
  if (lane == 0) ws[5L * NT + g] = wsum;
  __syncthreads();

  const int half = lane >> 4;
  const int n    = lane & 15;
  for (int c = 0; c < 5; ++c) {
    for (int mt = 0; mt < 10; ++mt) {
      const int m0 = mt * 16;
      v8f acc = box_tile(ch[c], m0);
      const int mb = m0 + half * 8;               // D rows: m = mb + j
#pragma unroll
      for (int j = 0; j < 8; ++j) outb[n * WSZ + mb + j] = acc[j];
    }
    __syncthreads();                              // LDS writes visible to TDM
    // memory line n: Sc[slab, h0+n, 0..159]  (160 contiguous, 16 lines, stride 160)
    tdm_store_2d(outb, ws + (long)c * NT + base, WSZ, 16, WSZ);
    __builtin_amdgcn_s_wait_tensorcnt(0);         // outb reusable next channel
    __syncthreads();
  }
}

// ---- Pass 2: H-axis box sum, in place via LDS staging -------------------
// 5 channels x 3200 groups (fixed slab, 16 consecutive w), 4 waves/block.
__global__ void __launch_bounds__(128) k_hpass(float* __restrict__ ws) {
  __shared__ float tin [4][2560];                 // tin[h*16 + p]
  __shared__ float tout[4][2560];                 // tout[m*16 + n]
  const int wid  = threadIdx.x >> 5;
  const int lane = threadIdx.x & 31;
  const int g    = blockIdx.x * 4 + wid;          // 0..15999
  const int c    = g / 3200;
  const int r    = g % 3200;
  const int slab = r / 10;
  const int w0   = (r % 10) * 16;
  float* Sc = ws + (long)c * NT;
  const long base = (long)slab * HW + w0;
  float* t = tin[wid];
  float* o = tout[wid];

  // 16-elem contiguous lines, 160 lines (h), line stride 160 elems
  tdm_load_2d(t, Sc + base, 16, WSZ, WSZ);
  __builtin_amdgcn_s_wait_tensorcnt(0);
  __syncthreads();

  const int half = lane >> 4;
  const int n    = lane & 15;
  for (int mt = 0; mt < 10; ++mt) {
    const int m0 = mt * 16;
    v8f acc = box_tile(t, m0);
    const int mb = m0 + half * 8;
#pragma unroll
    for (int j = 0; j < 8; ++j) o[(mb + j) * 16 + n] = acc[j];
  }
  __syncthreads();
  tdm_store_2d(o, Sc + base, 16, WSZ, WSZ);
  __builtin_amdgcn_s_wait_tensorcnt(0);           // complete before wave exit
}

// ---- Pass 3: D-axis box sum of all 5 channels + fused SSIM map ----------
// group = fixed (b,h), 16 consecutive w; 3200 single-wave blocks (50KB LDS).
__global__ void __launch_bounds__(32) k_dpass(float* __restrict__ ws) {
  __shared__ float tile[5][2560];                 // tile[c][d*16 + p]
  const int lane = threadIdx.x & 31;
  const int g  = blockIdx.x;                      // 0..3199
  const int b  = g / 1600;
  const int r  = g % 1600;
  const int h  = r / 10;
  const int w0 = (r % 10) * 16;
  const long base = (long)b * BVOL + (long)h * WSZ + w0;

  for (int c = 0; c < 5; ++c)                     // 5 async DMA tile loads
    tdm_load_2d(tile[c], ws + (long)c * NT + base, 16, WSZ, HW);
  __builtin_amdgcn_s_wait_tensorcnt(0);
  __syncthreads();

  const float inv = 1.0f / 1331.0f;               // 1 / 11^3
  const float c1  = 0.01f / (4096.0f * 4096.0f);  // faithful: K1 / range^2
  const float c2  = 0.03f / (4096.0f * 4096.0f);
  float ssum = 0.f;
  for (int mt = 0; mt < 10; ++mt) {
    const int m0 = mt * 16;
    v8f s0 = box_tile(tile[0], m0);
    v8f s1 = box_tile(tile[1], m0);
    v8f s2 = box_tile(tile[2], m0);
    v8f s3 = box_tile(tile[3], m0);
    v8f s4 = box_tile(tile[4], m0);
#pragma unroll
    for (int j = 0; j < 8; ++j) {
      const float mux = s0[j] * inv, muy = s1[j] * inv;
      const float ex2 = s2[j] * inv, ey2 = s3[j] * inv, exy = s4[j] * inv;
      const float mux2 = mux * mux, muy2 = muy * muy, muxy = mux * muy;
      const float sxx = ex2 - mux2, syy = ey2 - muy2, sxy = exy - muxy;
      const float nmr = (2.f * muxy + c1) * (2.f * sxy + c2);
      const float dnm = (mux2 + muy2 + c1) * (sxx + syy + c2);
      const float ss  = nmr / (dnm + 1e-8f);
      float v = 0.5f * (1.f - ss);
      v = v < 0.f ? 0.f : (v > 1.f ? 1.f : v);
      ssum += v;
    }
  }
  const float tot = wave_sum(ssum);
  if (lane == 0) ws[5L * NT + NPART + g] = tot;
}

// ---- Final deterministic reduction --------------------------------------
__global__ void __launch_bounds__(256) k_final(const float* __restrict__ part,
                                               float* __restrict__ out) {
  __shared__ float red[256];
  float l1 = 0.f, ss = 0.f;
  for (int i = threadIdx.x; i < NPART; i += 256) {
    l1 += part[i];
    ss += part[NPART + i];
  }
  red[threadIdx.x] = l1;
  __syncthreads();
  for (int s = 128; s > 0; s >>= 1) {
    if (threadIdx.x < s) red[threadIdx.x] += red[threadIdx.x + s];
    __syncthreads();
  }
  const float l1t = red[0];
  __syncthreads();
  red[threadIdx.x] = ss;
  __syncthreads();
  for (int s = 128; s > 0; s >>= 1) {
    if (threadIdx.x < s) red[threadIdx.x] += red[threadIdx.x + s];
    __syncthreads();
  }
  if (threadIdx.x == 0) {
    const float ssim_mean = red[0] / (float)NT;
    const float l1_mean   = l1t    / (float)NT;
    out[0] = 0.85f * ssim_mean + 0.15f * l1_mean;
  }
}

extern "C" void kernel_launch(void* const* d_in, const int* in_sizes, int n_in,
                              void* d_out, int out_size, void* d_ws, size_t ws_size,
                              hipStream_t stream) {
  (void)in_sizes; (void)n_in; (void)out_size; (void)ws_size;
  const float* x = (const float*)d_in[0];   // pred
  const float* y = (const float*)d_in[1];   // target
  float* ws  = (float*)d_ws;                // needs 5*NT floats + partials (~164 MB)
  float* out = (float*)d_out;

  k_wpass<<<3200, 32, 0, stream>>>(x, y, ws);   // channels + W box + L1 partials
  k_hpass<<<4000, 128, 0, stream>>>(ws);        // 16000 waves = 5ch * 3200 tiles
  k_dpass<<<3200, 32, 0, stream>>>(ws);         // D box + SSIM map + partials
  k_final<<<1, 256, 0, stream>>>(ws + 5L * NT, out);
}